// VectorizedPatchfier_17317308137721
// MI455X (gfx1250) — compile-verified
//
#include <hip/hip_runtime.h>

// ---------------------------------------------------------------------------
// VectorizedPatchfier for MI455X (gfx1250, wave32).
//
// Pipeline:
//   k_zero    : zero out_events + out_mask
//   k_hist    : per-chunk (2048 events) 16-bin patch histograms -> f32 (exact)
//   k_scan    : exclusive prefix sum across chunks per (batch,patch) via
//               v_wmma_f32_16x16x4_f32 (prefix-sum == Lstrict x H matmul;
//               all values < 2^24 so f32 WMMA is exact)
//   k_scatter : one wave per chunk, in-order 32-lane steps; stable rank via
//               4 bit-plane ballots + lane-owned counters + ds_bpermute shfl
// ---------------------------------------------------------------------------

typedef __attribute__((ext_vector_type(8))) float v8f;
typedef __attribute__((ext_vector_type(2))) float v2f;

#define CHUNK 2048          // events per wave-chunk (divides S = 1048576)
#define NPATCH 16           // 4x4 grid
#define INV_PATCH (1.0f / 128.0f)   // PATCH_W == PATCH_H == 128 (exact pow2)

// ---------------- zero output ----------------
__global__ void k_zero(float4* __restrict__ out4, long n4) {
  long i = (long)blockIdx.x * blockDim.x + threadIdx.x;
  long stride = (long)gridDim.x * blockDim.x;
  float4 z; z.x = 0.f; z.y = 0.f; z.z = 0.f; z.w = 0.f;
  for (; i < n4; i += stride) out4[i] = z;
}

__device__ __forceinline__ int patch_id(float x, float y) {
  int px = (int)floorf(x * INV_PATCH); px = px < 0 ? 0 : (px > 3 ? 3 : px);
  int py = (int)floorf(y * INV_PATCH); py = py < 0 ? 0 : (py > 3 ? 3 : py);
  return py * 4 + px;
}

// ---------------- per-chunk histogram ----------------
// grid = numChunks blocks, 256 threads. hist[chunk][16] as float (exact ints).
__global__ void k_hist(const float2* __restrict__ xy, float* __restrict__ hist) {
  __shared__ unsigned h[NPATCH];
  int t = threadIdx.x;
  if (t < NPATCH) h[t] = 0u;
  __syncthreads();
  long base = (long)blockIdx.x * CHUNK;
  for (int k = 0; k < CHUNK; k += 256) {
    long idx = base + k + t;
    if (k + 1024 < CHUNK)
      __builtin_prefetch(&xy[(idx + 1024) * 2], 0, 0);   // global_prefetch_b8
    float2 e = xy[idx * 2];            // (x,y) of event idx (stride 16B)
    atomicAdd(&h[patch_id(e.x, e.y)], 1u);
  }
  __syncthreads();
  if (t < NPATCH) hist[(long)blockIdx.x * NPATCH + t] = (float)h[t];
}

// ---------------- WMMA prefix scan ----------------
// One wave per batch. Exclusive scan over cpb (=512) chunks x 16 patches:
//   Base = Lstrict(cpb x cpb) * H(cpb x 16), tiled 16x16, K split into 4x(K=4).
// Running tile sum R accumulates earlier tiles; AllOnes*R gives their column
// sums (total earlier-chunk counts). Layouts per CDNA5 ISA 7.12.2:
//   A 16x4 : lane m (both halves M=lane&15); VGPR0=K(hi?2:0), VGPR1=K(hi?3:1)
//   B 4x16 : lane n = lane&15;               VGPR0=K(hi?2:0), VGPR1=K(hi?3:1)
//   C 16x16: VGPR v -> row v + (hi?8:0), col lane&15
__global__ void k_scan(const float* __restrict__ hist,
                       float* __restrict__ base_out, int cpb) {
  int lane = threadIdx.x & 31;
  int b    = threadIdx.x >> 5;         // batch handled by this wave
  int hi   = lane >> 4;
  int n    = lane & 15;                // col (patch) for B/C, row (M) for A
  int kx   = hi ? 2 : 0;
  int ky   = hi ? 3 : 1;

  v2f ones; ones.x = 1.0f; ones.y = 1.0f;
  v2f aL[4];
#pragma unroll
  for (int kk = 0; kk < 4; ++kk) {     // strict lower triangular columns
    aL[kk].x = (4 * kk + kx) < n ? 1.0f : 0.0f;
    aL[kk].y = (4 * kk + ky) < n ? 1.0f : 0.0f;
  }

  v2f R[4];
#pragma unroll
  for (int kk = 0; kk < 4; ++kk) { R[kk].x = 0.0f; R[kk].y = 0.0f; }

  int gb = b * cpb;                    // first chunk of this batch
  int ntiles = cpb >> 4;               // 32
  for (int I = 0; I < ntiles; ++I) {
    int rowbase = gb + I * 16;
    v2f Hb[4];
#pragma unroll
    for (int kk = 0; kk < 4; ++kk) {
      Hb[kk].x = hist[(long)(rowbase + 4 * kk + kx) * NPATCH + n];
      Hb[kk].y = hist[(long)(rowbase + 4 * kk + ky) * NPATCH + n];
    }
    v8f c = {};
#pragma unroll
    for (int kk = 0; kk < 4; ++kk)     // column sums of all earlier tiles
      c = __builtin_amdgcn_wmma_f32_16x16x4_f32(false, ones, false, R[kk],
                                                (short)0, c, false, false);
#pragma unroll
    for (int kk = 0; kk < 4; ++kk)     // strict-lower within this tile
      c = __builtin_amdgcn_wmma_f32_16x16x4_f32(false, aL[kk], false, Hb[kk],
                                                (short)0, c, false, false);
#pragma unroll
    for (int v = 0; v < 8; ++v) {
      int row = rowbase + v + hi * 8;
      base_out[(long)row * NPATCH + n] = c[v];
    }
#pragma unroll
    for (int kk = 0; kk < 4; ++kk) { R[kk].x += Hb[kk].x; R[kk].y += Hb[kk].y; }
  }
}

// ---------------- stable scatter ----------------
// One wave per chunk; serial 32-event steps keep original order.
// Stable rank within a step: 4 bit-plane ballots reconstruct the same-pid
// lane mask (AND of selected planes). Lane p (<16) owns the running counter
// for patch p; the pre-update base is fetched with __shfl (ds_bpermute),
// and lane p updates its counter from the reconstructed mask for patch p.
__global__ void k_scatter(const float4* __restrict__ ev4,
                          const float* __restrict__ base_in,
                          const int* __restrict__ pL,
                          float4* __restrict__ out_ev,
                          float* __restrict__ out_mask, int cpb) {
  int lane  = threadIdx.x & 31;
  int wave  = threadIdx.x >> 5;
  int chunk = blockIdx.x * (blockDim.x >> 5) + wave;
  int L = pL[0];

  int cnt = 0x7fffffff;                // lane p < 16 owns counter for patch p
  if (lane < NPATCH) cnt = (int)base_in[(long)chunk * NPATCH + lane];
  unsigned roomb =
      __builtin_amdgcn_ballot_w32(lane < NPATCH && cnt < L);
  if (roomb == 0u) return;             // wave-uniform: chunk cannot contribute

  int batch = chunk / cpb;
  long ebase = (long)chunk * CHUNK;
  unsigned lt = (1u << lane) - 1u;

  for (int s = 0; s < CHUNK / 32; ++s) {
    if (s + 8 < CHUNK / 32)
      __builtin_prefetch(&ev4[ebase + (long)(s + 8) * 32 + lane], 0, 0);
    float4 e = ev4[ebase + (long)s * 32 + lane];
    int pid = patch_id(e.x, e.y);

    // 4 bit-plane ballots (wave-uniform values)
    unsigned b0 = __builtin_amdgcn_ballot_w32((pid & 1) != 0);
    unsigned b1 = __builtin_amdgcn_ballot_w32((pid & 2) != 0);
    unsigned b2 = __builtin_amdgcn_ballot_w32((pid & 4) != 0);
    unsigned b3 = __builtin_amdgcn_ballot_w32((pid & 8) != 0);

    // mask of lanes whose pid equals mine
    unsigned m = ((pid & 1) ? b0 : ~b0) & ((pid & 2) ? b1 : ~b1) &
                 ((pid & 4) ? b2 : ~b2) & ((pid & 8) ? b3 : ~b3);
    int before = __popc(m & lt);
    int basec  = __shfl(cnt, pid, 32); // pre-update counter of patch pid

    // lane p updates counter for patch p
    unsigned mp = ((lane & 1) ? b0 : ~b0) & ((lane & 2) ? b1 : ~b1) &
                  ((lane & 4) ? b2 : ~b2) & ((lane & 8) ? b3 : ~b3);
    if (lane < NPATCH) cnt += __popc(mp);

    int rank = basec + before;
    if (rank < L) {
      int dest = (batch * NPATCH + pid) * L + rank;
      float4 o;
      o.x = e.x - (float)((pid & 3) * 128);
      o.y = e.y - (float)((pid >> 2) * 128);
      o.z = e.z; o.w = e.w;
      out_ev[dest]   = o;
      out_mask[dest] = 1.0f;
    }
    if ((s & 15) == 15) {              // all groups full -> stop reading
      unsigned r2 = __builtin_amdgcn_ballot_w32(lane < NPATCH && cnt < L);
      if (r2 == 0u) break;
    }
  }
}

extern "C" void kernel_launch(void* const* d_in, const int* in_sizes, int n_in,
                              void* d_out, int out_size, void* d_ws, size_t ws_size,
                              hipStream_t stream) {
  const float4* ev4 = (const float4*)d_in[0];
  const float2* xy  = (const float2*)d_in[0];
  const int*    pL  = (const int*)d_in[1];

  long n_elems  = (long)in_sizes[0];       // B*S*C = 33,554,432
  long n_events = n_elems / 4;             // 8,388,608
  int  chunks   = (int)(n_events / CHUNK); // 4096
  const int B   = 8;                       // module hyperparameter
  int  cpb      = chunks / B;              // 512 chunks per batch

  // out = [B,16,L,4] events followed by [B,16,L] mask, both float32
  long n_ev_floats = (long)out_size / 5 * 4;     // 4,194,304
  float*  out      = (float*)d_out;
  float4* out_ev   = (float4*)out;
  float*  out_mask = out + n_ev_floats;

  float* hist = (float*)d_ws;                    // [chunks][16]
  float* base = hist + (long)chunks * NPATCH;    // [chunks][16]  (512 KB total)

  long n4 = (long)out_size / 4;                  // whole output is 16B-multiple
  k_zero<<<2048, 256, 0, stream>>>((float4*)out, n4);
  k_hist<<<chunks, 256, 0, stream>>>(xy, hist);
  k_scan<<<1, 32 * B, 0, stream>>>(hist, base, cpb);
  k_scatter<<<chunks / 8, 256, 0, stream>>>(ev4, base, pL, out_ev, out_mask, cpb);
}